// attn_model_ad_38268158607550
// MI455X (gfx1250) — compile-verified
//
#include <hip/hip_runtime.h>
#include <math.h>

// Problem dims (fixed by reference)
#define B_  16
#define T_  32
#define S_  128
#define F_  3
#define H_  256
#define NH_ 8
#define DK_ 32
#define TP_ 4
#define M_  (B_*S_)     // 2048 rows = B*S
#define KE_ 544         // enc concat K: 3+256+256=515 padded to mult of 32
#define KD_ 768         // dec concat K: 256*3
#define G4_ 1024        // 4*H gate width
#define KS_ 32          // K step per LDS stage

typedef float v2f __attribute__((ext_vector_type(2)));
typedef float v8f __attribute__((ext_vector_type(8)));

__device__ __forceinline__ float sigmoidf_(float x) { return 1.0f / (1.0f + __expf(-x)); }

// ---------------------------------------------------------------------------
// FP32 GEMM via V_WMMA_F32_16X16X4_F32, double-buffered LDS, K-stage = 32.
// C[M,N] = A[M,K] * W[K,N] (+bias) (+optional sigmoid). M,N mult of 64, K mult of 32.
// Block: 256 threads = 8 waves; block tile 64x64; wave tile 16x32 (two 16x16 accs).
// LDS layout: As[buf][row][k] and Bs[buf][col][k], both row-stride 36 so a lane's
// K-pair is one aligned ds_load_b64 and both 16-lane halves are bank-conflict-free.
// A frag (16x4 f32): lane<16 -> row=lane, K pair {0,1}; lane>=16 -> row=lane-16, {2,3}.
// B frag (4x16 f32): lane&15 -> col, same K-pair split by lane>>4.
// C/D (16x16 f32, 8 VGPRs): vgpr r -> M = r + 8*(lane>>4), N = lane&15.
// ---------------------------------------------------------------------------
__global__ __launch_bounds__(256)
void gemm_wmma(const float* __restrict__ A, const float* __restrict__ W,
               const float* __restrict__ bias, float* __restrict__ C,
               int N, int K, int act)
{
    __shared__ float As[2][64][36];   // 2 x 9216 B
    __shared__ float Bs[2][64][36];   // transposed: Bs[buf][n][k]

    const int tid  = threadIdx.x;
    const int lane = tid & 31;
    const int wave = tid >> 5;
    const int wr   = wave & 3;      // 4 row-tiles of 16
    const int wc   = wave >> 2;     // 2 col-tiles of 32
    const int m0   = blockIdx.x * 64;
    const int n0   = blockIdx.y * 64;

    v8f acc0 = {0.f,0.f,0.f,0.f,0.f,0.f,0.f,0.f};
    v8f acc1 = acc0;

    // cooperative-load coordinates
    const int ar = tid >> 2, aq = (tid & 3) << 2;   // A: 64 rows x 2 float4 (k and k+16)
    const int bk = tid >> 3, bn = (tid & 7) << 2;   // B: 32 k-rows x 2 float4 (n and n+32)
    const int lm = lane & 15;
    const int kh = (lane >> 4) << 1;                // 0 or 2

    const int nstages = K / KS_;

    // ---- preload stage 0 ----
    {
        float4 a0 = *reinterpret_cast<const float4*>(A + (size_t)(m0 + ar) * K + aq);
        float4 a1 = *reinterpret_cast<const float4*>(A + (size_t)(m0 + ar) * K + aq + 16);
        float4 b0 = *reinterpret_cast<const float4*>(W + (size_t)bk * N + n0 + bn);
        float4 b1 = *reinterpret_cast<const float4*>(W + (size_t)bk * N + n0 + bn + 32);
        *reinterpret_cast<float4*>(&As[0][ar][aq])      = a0;
        *reinterpret_cast<float4*>(&As[0][ar][aq + 16]) = a1;
        Bs[0][bn + 0][bk] = b0.x;  Bs[0][bn + 1][bk] = b0.y;
        Bs[0][bn + 2][bk] = b0.z;  Bs[0][bn + 3][bk] = b0.w;
        Bs[0][bn + 32][bk] = b1.x; Bs[0][bn + 33][bk] = b1.y;
        Bs[0][bn + 34][bk] = b1.z; Bs[0][bn + 35][bk] = b1.w;
    }
    __syncthreads();

    for (int s = 0; s < nstages; ++s) {
        const int cur  = s & 1;
        const int nxt  = cur ^ 1;
        const bool more = (s + 1 < nstages);

        float4 a0, a1, b0, b1;
        if (more) {
            const int k0 = (s + 1) * KS_;
            a0 = *reinterpret_cast<const float4*>(A + (size_t)(m0 + ar) * K + k0 + aq);
            a1 = *reinterpret_cast<const float4*>(A + (size_t)(m0 + ar) * K + k0 + aq + 16);
            b0 = *reinterpret_cast<const float4*>(W + (size_t)(k0 + bk) * N + n0 + bn);
            b1 = *reinterpret_cast<const float4*>(W + (size_t)(k0 + bk) * N + n0 + bn + 32);
            if (s + 2 < nstages) {
                const int k2 = (s + 2) * KS_;
                __builtin_prefetch(A + (size_t)(m0 + ar) * K + k2 + aq, 0, 1);
                __builtin_prefetch(W + (size_t)(k2 + bk) * N + n0 + bn, 0, 1);
            }
        }

#pragma unroll
        for (int kk = 0; kk < KS_; kk += 4) {
            v2f a   = *reinterpret_cast<const v2f*>(&As[cur][wr * 16 + lm][kk + kh]);
            v2f bb0 = *reinterpret_cast<const v2f*>(&Bs[cur][wc * 32 + lm][kk + kh]);
            v2f bb1 = *reinterpret_cast<const v2f*>(&Bs[cur][wc * 32 + 16 + lm][kk + kh]);
            acc0 = __builtin_amdgcn_wmma_f32_16x16x4_f32(false, a, false, bb0, (short)0, acc0, false, false);
            acc1 = __builtin_amdgcn_wmma_f32_16x16x4_f32(false, a, false, bb1, (short)0, acc1, false, false);
        }

        if (more) {
            *reinterpret_cast<float4*>(&As[nxt][ar][aq])      = a0;
            *reinterpret_cast<float4*>(&As[nxt][ar][aq + 16]) = a1;
            Bs[nxt][bn + 0][bk] = b0.x;  Bs[nxt][bn + 1][bk] = b0.y;
            Bs[nxt][bn + 2][bk] = b0.z;  Bs[nxt][bn + 3][bk] = b0.w;
            Bs[nxt][bn + 32][bk] = b1.x; Bs[nxt][bn + 33][bk] = b1.y;
            Bs[nxt][bn + 34][bk] = b1.z; Bs[nxt][bn + 35][bk] = b1.w;
            __syncthreads();
        }
    }

    const int cn = n0 + wc * 32 + lm;
    const int cm = m0 + wr * 16 + (lane >> 4) * 8;
    const float b0v = bias ? bias[cn]      : 0.0f;
    const float b1v = bias ? bias[cn + 16] : 0.0f;
#pragma unroll
    for (int r = 0; r < 8; ++r) {
        float v0 = acc0[r] + b0v;
        float v1 = acc1[r] + b1v;
        if (act == 1) { v0 = sigmoidf_(v0); v1 = sigmoidf_(v1); }
        C[(size_t)(cm + r) * N + cn]      = v0;
        C[(size_t)(cm + r) * N + cn + 16] = v1;
    }
}

// ---------------------------------------------------------------------------
// 4-neighbor attention. sp[n]@Wk == row-shift of (h@Wk), so we gather shifted
// rows of kk/vv. OOB neighbors contribute score 0 (exp(0) in softmax) and v=0,
// exactly matching the zero-padded reference. One thread per (b,s,head).
// ---------------------------------------------------------------------------
__global__ __launch_bounds__(256)
void attn_kernel(const float* __restrict__ q, const float* __restrict__ kk,
                 const float* __restrict__ vv, float* __restrict__ ctx)
{
    int g = blockIdx.x * blockDim.x + threadIdx.x;
    if (g >= M_ * NH_) return;
    const int head = g & (NH_ - 1);
    const int bs   = g >> 3;
    const int s    = bs & (S_ - 1);
    const int offs[4] = {2, 1, -1, -2};   // after2, after1, before1, before2

    const float* qrow = q + (size_t)bs * H_ + head * DK_;
    float qv[DK_];
#pragma unroll
    for (int d = 0; d < DK_; ++d) qv[d] = qrow[d];

    float sc[4];
    int   rr[4];
    int   inb[4];
#pragma unroll
    for (int n = 0; n < 4; ++n) {
        const int r = s + offs[n];
        inb[n] = (r >= 0 && r < S_);
        rr[n]  = bs + offs[n];
        float dot = 0.f;
        if (inb[n]) {
            const float* krow = kk + (size_t)rr[n] * H_ + head * DK_;
#pragma unroll
            for (int d = 0; d < DK_; ++d) dot += qv[d] * krow[d];
        }
        sc[n] = dot * 0.17677669529663687f;   // 1/sqrt(32)
    }
    const float mx = fmaxf(fmaxf(sc[0], sc[1]), fmaxf(sc[2], sc[3]));
    float e[4], ssum = 0.f;
#pragma unroll
    for (int n = 0; n < 4; ++n) { e[n] = __expf(sc[n] - mx); ssum += e[n]; }
    const float inv = 1.0f / ssum;

    float outv[DK_];
#pragma unroll
    for (int d = 0; d < DK_; ++d) outv[d] = 0.f;
#pragma unroll
    for (int n = 0; n < 4; ++n) {
        if (!inb[n]) continue;
        const float w = e[n] * inv;
        const float* vrow = vv + (size_t)rr[n] * H_ + head * DK_;
#pragma unroll
        for (int d = 0; d < DK_; ++d) outv[d] += w * vrow[d];
    }
    float* crow = ctx + (size_t)bs * H_ + head * DK_;
#pragma unroll
    for (int d = 0; d < DK_; ++d) crow[d] = outv[d];
}

// Pack [x(3) | h(256) | ctx(256) | zero-pad] -> xcat[2048][544]
__global__ void pack_enc(const float* __restrict__ xin, int t,
                         const float* __restrict__ eh, const float* __restrict__ ctx,
                         float* __restrict__ xcat)
{
    int g = blockIdx.x * blockDim.x + threadIdx.x;
    if (g >= M_ * KE_) return;
    const int col = g % KE_;
    const int bs  = g / KE_;
    float v;
    if (col < F_) {
        const int b = bs >> 7, s = bs & (S_ - 1);
        v = xin[(((size_t)b * T_ + t) * S_ + s) * F_ + col];
    } else if (col < F_ + H_) {
        v = eh[(size_t)bs * H_ + (col - F_)];
    } else if (col < F_ + 2 * H_) {
        v = ctx[(size_t)bs * H_ + (col - F_ - H_)];
    } else {
        v = 0.f;
    }
    xcat[g] = v;
}

// Pack [code(256) | h(256) | ctx(256)] -> xcat[2048][768]
__global__ void pack_dec(const float* __restrict__ code, const float* __restrict__ dh,
                         const float* __restrict__ ctx, float* __restrict__ xcat)
{
    int g = blockIdx.x * blockDim.x + threadIdx.x;
    if (g >= M_ * KD_) return;
    const int col = g % KD_;
    const int bs  = g / KD_;
    float v;
    if (col < H_)           v = code[(size_t)bs * H_ + col];
    else if (col < 2 * H_)  v = dh[(size_t)bs * H_ + (col - H_)];
    else                    v = ctx[(size_t)bs * H_ + (col - 2 * H_)];
    xcat[g] = v;
}

// Zero-pad enc_Wg [515][1024] -> [544][1024]
__global__ void pad_wg(const float* __restrict__ Wg, float* __restrict__ Wp)
{
    int g = blockIdx.x * blockDim.x + threadIdx.x;
    if (g >= KE_ * G4_) return;
    const int row = g / G4_;
    Wp[g] = (row < F_ + 2 * H_) ? Wg[(size_t)row * G4_ + (g % G4_)] : 0.f;
}

// LSTM gating. z split i|f|g|o along 1024 cols. In-place safe (per-thread RAW).
__global__ void gates_kernel(const float* z, const float* c_in,
                             float* h_out, float* c_out)
{
    int g = blockIdx.x * blockDim.x + threadIdx.x;
    if (g >= M_ * H_) return;
    const int row = g >> 8, j = g & (H_ - 1);
    const float* zr = z + (size_t)row * G4_;
    const float iv = zr[j], fv = zr[j + H_], gv = zr[j + 2 * H_], ov = zr[j + 3 * H_];
    const float cn = sigmoidf_(fv) * c_in[g] + sigmoidf_(iv) * tanhf(gv);
    const float hn = sigmoidf_(ov) * tanhf(cn);
    c_out[g] = cn;
    h_out[g] = hn;
}

// out = dh@outW + b; In[s]=out[s-1] (In[0]=x[t+1,0,1]); num = x[t,s,2]+In-out.
// One thread per (b,s); recompute neighbor dot (cheap) to avoid a 2nd pass.
__global__ void output_kernel(const float* __restrict__ dh, const float* __restrict__ outW,
                              const float* __restrict__ outB, const float* __restrict__ xin,
                              int t, float* __restrict__ y)
{
    int g = blockIdx.x * blockDim.x + threadIdx.x;
    if (g >= M_) return;
    const int b = g >> 7, s = g & (S_ - 1);
    const float ob = outB[0];
    float o = ob;
    const float* hr = dh + (size_t)g * H_;
    for (int k = 0; k < H_; ++k) o += hr[k] * outW[k];
    float In;
    if (s == 0) {
        In = xin[(((size_t)b * T_ + (t + 1)) * S_ + 0) * F_ + 1];
    } else {
        const float* hp = dh + (size_t)(g - 1) * H_;
        float op = ob;
        for (int k = 0; k < H_; ++k) op += hp[k] * outW[k];
        In = op;
    }
    const float xc2 = xin[(((size_t)b * T_ + t) * S_ + s) * F_ + 2];
    const float num = xc2 + In - o;
    const int tt = t - TP_;
    const size_t base = (((size_t)b * (T_ - TP_ - 1) + tt) * S_ + s) * 3;
    y[base + 0] = o;
    y[base + 1] = In;
    y[base + 2] = num;
}

// ---------------------------------------------------------------------------
extern "C" void kernel_launch(void* const* d_in, const int* in_sizes, int n_in,
                              void* d_out, int out_size, void* d_ws, size_t ws_size,
                              hipStream_t stream)
{
    (void)in_sizes; (void)n_in; (void)out_size; (void)ws_size;
    const float* xin  = (const float*)d_in[0];
    const float* eWq  = (const float*)d_in[1];
    const float* eWk  = (const float*)d_in[2];
    const float* eWv  = (const float*)d_in[3];
    const float* eWg  = (const float*)d_in[4];
    const float* ebg  = (const float*)d_in[5];
    const float* dWq  = (const float*)d_in[6];
    const float* dWk  = (const float*)d_in[7];
    const float* dWv  = (const float*)d_in[8];
    const float* dWg  = (const float*)d_in[9];
    const float* dbg  = (const float*)d_in[10];
    const float* embW = (const float*)d_in[11];
    const float* embB = (const float*)d_in[12];
    const float* outW = (const float*)d_in[13];
    const float* outB = (const float*)d_in[14];
    float* y = (float*)d_out;

    // Workspace layout (floats); total ~36 MB
    float* ws = (float*)d_ws;
    const size_t SZ = (size_t)M_ * H_;     // 2048*256
    float* eh   = ws + 0 * SZ;
    float* ec   = ws + 1 * SZ;
    float* dh   = ws + 2 * SZ;
    float* qb   = ws + 3 * SZ;
    float* kb   = ws + 4 * SZ;
    float* vb   = ws + 5 * SZ;
    float* ctx  = ws + 6 * SZ;
    float* code = ws + 7 * SZ;
    float* cD   = ws + 8 * SZ;             // decoder c_new (discarded)
    float* zb   = ws + 9 * SZ;             // 2048 x 1024
    float* xcat = ws + 13 * SZ;            // 2048 x 768 (enc uses stride 544)
    float* wgp  = ws + 16 * SZ;            // 544 x 1024 padded enc_Wg

    hipMemsetAsync(eh, 0, 3 * SZ * sizeof(float), stream);   // eh, ec, dh -> 0

    dim3 blk(256);
    pad_wg<<<(KE_ * G4_ + 255) / 256, blk, 0, stream>>>(eWg, wgp);

    const dim3 gq(M_ / 64, H_ / 64);    // 32 x 4 : [2048,256,256]
    const dim3 gg(M_ / 64, G4_ / 64);   // 32 x 16: [2048,1024,K]

    for (int t = 0; t < T_ - 1; ++t) {
        // ---- encoder cell ----
        gemm_wmma<<<gq, blk, 0, stream>>>(eh, eWq, nullptr, qb, H_, H_, 0);
        gemm_wmma<<<gq, blk, 0, stream>>>(eh, eWk, nullptr, kb, H_, H_, 0);
        gemm_wmma<<<gq, blk, 0, stream>>>(eh, eWv, nullptr, vb, H_, H_, 0);
        attn_kernel<<<(M_ * NH_ + 255) / 256, blk, 0, stream>>>(qb, kb, vb, ctx);
        pack_enc<<<(M_ * KE_ + 255) / 256, blk, 0, stream>>>(xin, t, eh, ctx, xcat);
        gemm_wmma<<<gg, blk, 0, stream>>>(xcat, wgp, ebg, zb, G4_, KE_, 0);
        gates_kernel<<<(M_ * H_ + 255) / 256, blk, 0, stream>>>(zb, ec, eh, ec);
        // ---- embedding ----
        gemm_wmma<<<gq, blk, 0, stream>>>(eh, embW, embB, code, H_, H_, 1);
        // ---- decoder cell (c := dh_old) ----
        gemm_wmma<<<gq, blk, 0, stream>>>(dh, dWq, nullptr, qb, H_, H_, 0);
        gemm_wmma<<<gq, blk, 0, stream>>>(dh, dWk, nullptr, kb, H_, H_, 0);
        gemm_wmma<<<gq, blk, 0, stream>>>(dh, dWv, nullptr, vb, H_, H_, 0);
        attn_kernel<<<(M_ * NH_ + 255) / 256, blk, 0, stream>>>(qb, kb, vb, ctx);
        pack_dec<<<(M_ * KD_ + 255) / 256, blk, 0, stream>>>(code, dh, ctx, xcat);
        gemm_wmma<<<gg, blk, 0, stream>>>(xcat, dWg, dbg, zb, G4_, KD_, 0);
        gates_kernel<<<(M_ * H_ + 255) / 256, blk, 0, stream>>>(zb, dh, dh, cD);
        // ---- output head ----
        if (t >= TP_) {
            output_kernel<<<(M_ + 255) / 256, blk, 0, stream>>>(dh, outW, outB, xin, t, y);
        }
    }
}